// GNN_60421599920514
// MI455X (gfx1250) — compile-verified
//
#include <hip/hip_runtime.h>

#define N_NODES_C  100000
#define N_GRAPHS_C 512
#define HID_C      64
#define IN_DIM_C   8
#define OUT_DIM_C  2

typedef __attribute__((ext_vector_type(2))) float v2f;
typedef __attribute__((ext_vector_type(8))) float v8f;

// ---------------------------------------------------------------- zero ----
__global__ void gnn_zero_kernel(float* __restrict__ p, int n) {
  int i = blockIdx.x * blockDim.x + threadIdx.x;
  if (i < n) p[i] = 0.0f;
}

// ------------------------------------------------------------- scatter ----
// One thread per (edge, feature). Consecutive threads share an edge and walk
// consecutive features -> coalesced gather loads and clustered f32 atomics
// that stay resident in the 192MB L2.
__global__ void gnn_scatter_kernel(const float* __restrict__ h,
                                   const int* __restrict__ src,
                                   const int* __restrict__ dst,
                                   float* __restrict__ agg,
                                   int nEdges, int kShift) {
  int t = blockIdx.x * blockDim.x + threadIdx.x;
  int e = t >> kShift;
  if (e >= nEdges) return;
  int K = 1 << kShift;
  int f = t & (K - 1);
  atomicAdd(&agg[(size_t)dst[e] * K + f], h[(size_t)src[e] * K + f]);
}

// ----------------------------------------------------------- WMMA GEMM ----
// out[m, 0:64] = act( agg[m, 0:K] @ Wrel + bias + root[m, 0:K] @ Wroot )
// One wave32 owns a 16(M) x 64(N) tile; K-loop in steps of 4 using
// v_wmma_f32_16x16x4_f32 (full fp32 precision).
//
// ISA fragment layouts (CDNA5 ISA):
//   A 16x4 f32 : lanes 0-15 -> {K=0,K=1}, lanes 16-31 -> {K=2,K=3}, M = lane%16
//   B 4x16 f32 : lanes 0-15 -> {K=0,K=1}, lanes 16-31 -> {K=2,K=3}, N = lane%16
//   C/D 16x16  : VGPR r -> row r (lanes 0-15) / row r+8 (lanes 16-31)
__global__ void gnn_gemm_wmma_kernel(const float* __restrict__ Aagg,
                                     const float* __restrict__ Aroot,
                                     const float* __restrict__ Wrel,
                                     const float* __restrict__ bias,
                                     const float* __restrict__ Wroot,
                                     float* __restrict__ out,
                                     int K, int doRelu) {
  const int wave = (blockIdx.x * blockDim.x + threadIdx.x) >> 5;
  const int lane = threadIdx.x & 31;
  const int m0 = wave * 16;
  if (m0 >= N_NODES_C) return;          // wave-uniform: EXEC stays all-ones
  const int half = lane >> 4;           // 0: K pair {0,1}, 1: K pair {2,3}
  const int lm   = lane & 15;           // A row / B,C column within tile
  const int arow = m0 + lm;

  v8f acc[4];
#pragma unroll
  for (int t = 0; t < 4; ++t)
    acc[t] = (v8f){0.f, 0.f, 0.f, 0.f, 0.f, 0.f, 0.f, 0.f};

  for (int k0 = 0; k0 < K; k0 += 4) {
    const int kb = k0 + 2 * half;
    // A fragments: contiguous float2 along K (8B-aligned: kb is even)
    const float2 fa = *reinterpret_cast<const float2*>(Aagg  + (size_t)arow * K + kb);
    const float2 fr = *reinterpret_cast<const float2*>(Aroot + (size_t)arow * K + kb);
    const v2f a_agg = { fa.x, fa.y };
    const v2f a_rt  = { fr.x, fr.y };
#pragma unroll
    for (int t = 0; t < 4; ++t) {
      const int col = t * 16 + lm;
      const v2f b_rel = { Wrel [kb * HID_C + col], Wrel [(kb + 1) * HID_C + col] };
      const v2f b_rt  = { Wroot[kb * HID_C + col], Wroot[(kb + 1) * HID_C + col] };
      acc[t] = __builtin_amdgcn_wmma_f32_16x16x4_f32(
          false, a_agg, false, b_rel, (short)0, acc[t], false, false);
      acc[t] = __builtin_amdgcn_wmma_f32_16x16x4_f32(
          false, a_rt,  false, b_rt,  (short)0, acc[t], false, false);
    }
  }

#pragma unroll
  for (int t = 0; t < 4; ++t) {
    const int col = t * 16 + lm;
    const float bv = bias[col];
#pragma unroll
    for (int r = 0; r < 8; ++r) {
      const int row = m0 + r + 8 * half;
      float v = acc[t][r] + bv;
      if (doRelu) v = fmaxf(v, 0.0f);
      out[(size_t)row * HID_C + col] = v;
    }
  }
}

// ---------------------------------------------------------------- pool ----
__global__ void gnn_pool_sum_kernel(const float* __restrict__ h,
                                    const int* __restrict__ batch,
                                    float* __restrict__ sums) {
  int t = blockIdx.x * blockDim.x + threadIdx.x;
  if (t >= N_NODES_C * HID_C) return;
  int n = t >> 6;
  int f = t & (HID_C - 1);
  atomicAdd(&sums[(size_t)batch[n] * HID_C + f], h[t]);
}

__global__ void gnn_pool_cnt_kernel(const int* __restrict__ batch,
                                    float* __restrict__ cnts) {
  int n = blockIdx.x * blockDim.x + threadIdx.x;
  if (n >= N_NODES_C) return;
  atomicAdd(&cnts[batch[n]], 1.0f);
}

// --------------------------------------------------------------- final ----
__global__ void gnn_final_kernel(const float* __restrict__ sums,
                                 const float* __restrict__ cnts,
                                 const float* __restrict__ w,
                                 const float* __restrict__ b,
                                 float* __restrict__ out) {
  int t = blockIdx.x * blockDim.x + threadIdx.x;
  if (t >= N_GRAPHS_C * OUT_DIM_C) return;
  int g = t / OUT_DIM_C;
  int o = t % OUT_DIM_C;
  float inv = 1.0f / fmaxf(cnts[g], 1.0f);
  float acc = b[o];
#pragma unroll
  for (int k = 0; k < HID_C; ++k)
    acc += (sums[(size_t)g * HID_C + k] * inv) * w[k * OUT_DIM_C + o];
  out[t] = acc;
}

// -------------------------------------------------------------- launch ----
extern "C" void kernel_launch(void* const* d_in, const int* in_sizes, int n_in,
                              void* d_out, int out_size, void* d_ws, size_t ws_size,
                              hipStream_t stream) {
  const float* x     = (const float*)d_in[0];
  const int*   ei    = (const int*)d_in[1];
  const int*   batch = (const int*)d_in[2];
  const int nE = in_sizes[1] / 2;
  const int* src = ei;
  const int* dst = ei + nE;

  const float* w_rel[4]  = { (const float*)d_in[3], (const float*)d_in[6],
                             (const float*)d_in[9], (const float*)d_in[12] };
  const float* b_rel[4]  = { (const float*)d_in[4], (const float*)d_in[7],
                             (const float*)d_in[10], (const float*)d_in[13] };
  const float* w_root[4] = { (const float*)d_in[5], (const float*)d_in[8],
                             (const float*)d_in[11], (const float*)d_in[14] };
  const float* w_lin = (const float*)d_in[15];
  const float* b_lin = (const float*)d_in[16];

  float* bufA = (float*)d_ws;                        // N_NODES * 64
  float* bufB = bufA + (size_t)N_NODES_C * HID_C;    // N_NODES * 64
  float* agg  = bufB + (size_t)N_NODES_C * HID_C;    // N_NODES * 64
  float* sums = agg  + (size_t)N_NODES_C * HID_C;    // 512 * 64
  float* cnts = sums + (size_t)N_GRAPHS_C * HID_C;   // 512 (contiguous w/ sums)

  const int B = 256;                                 // 8 wave32s per block
  const int waves = N_NODES_C / 16;                  // 6250, exact
  const int gemmBlocks = (waves * 32 + B - 1) / B;

  // ---- layer 1 (K = 8, input x) -> bufA
  {
    int nAgg = N_NODES_C * IN_DIM_C;
    gnn_zero_kernel<<<(nAgg + B - 1) / B, B, 0, stream>>>(agg, nAgg);
    int nT = nE * IN_DIM_C;
    gnn_scatter_kernel<<<(nT + B - 1) / B, B, 0, stream>>>(x, src, dst, agg, nE, 3);
    gnn_gemm_wmma_kernel<<<gemmBlocks, B, 0, stream>>>(
        agg, x, w_rel[0], b_rel[0], w_root[0], bufA, IN_DIM_C, 1);
  }

  // ---- layers 2..4 (K = 64), ping-pong bufA <-> bufB
  const float* cur = bufA;
  float* nxt = bufB;
  for (int L = 1; L < 4; ++L) {
    int nAgg = N_NODES_C * HID_C;
    gnn_zero_kernel<<<(nAgg + B - 1) / B, B, 0, stream>>>(agg, nAgg);
    int nT = nE * HID_C;                              // 64M threads
    gnn_scatter_kernel<<<(nT + B - 1) / B, B, 0, stream>>>(cur, src, dst, agg, nE, 6);
    gnn_gemm_wmma_kernel<<<gemmBlocks, B, 0, stream>>>(
        agg, cur, w_rel[L], b_rel[L], w_root[L], nxt, HID_C, (L < 3) ? 1 : 0);
    const float* tmp = cur; cur = nxt; nxt = (float*)tmp;
  }
  // after the loop `cur` holds the layer-4 output

  // ---- global mean pool + final linear
  {
    int nZ = N_GRAPHS_C * HID_C + N_GRAPHS_C;         // sums + cnts contiguous
    gnn_zero_kernel<<<(nZ + B - 1) / B, B, 0, stream>>>(sums, nZ);
    int nT = N_NODES_C * HID_C;
    gnn_pool_sum_kernel<<<(nT + B - 1) / B, B, 0, stream>>>(cur, batch, sums);
    gnn_pool_cnt_kernel<<<(N_NODES_C + B - 1) / B, B, 0, stream>>>(batch, cnts);
    gnn_final_kernel<<<(N_GRAPHS_C * OUT_DIM_C + B - 1) / B, B, 0, stream>>>(
        sums, cnts, w_lin, b_lin, (float*)d_out);
  }
}